// TranslateLayer_60885456388682
// MI455X (gfx1250) — compile-verified
//
#include <hip/hip_runtime.h>
#include <hip/hip_bf16.h>

typedef __attribute__((ext_vector_type(2))) float v2f;
typedef __attribute__((ext_vector_type(8))) float v8f;

// ---------------------------------------------------------------------------
// Problem constants
// ---------------------------------------------------------------------------
#define BB    8
#define CIMG  3
#define HH    512
#define WW    512
#define CELLN 16
#define NHW   32              // nh == nw == 32
#define CC    64              // ctrl channels
#define C1    50              // conv1 out channels
#define C2    9               // conv2 out channels
#define K1    (CC * 9)        // 576
#define NPOS  (NHW * NHW)     // 1024 spatial positions per batch

// ---------------------------------------------------------------------------
// Kernel 1: conv3x3 (64 -> 50) + bias + ReLU as implicit GEMM on
// V_WMMA_F32_16X16X4_F32.  One wave computes a 16(co) x 16(pos) output tile.
// K ordered as (tap, ci): k = tap*64 + ci, so the step-4 inner loop stays
// inside one tap (64 % 4 == 0) and needs no div/mod.
//
// No predicated loads:
//  - A rows m >= 50 produce garbage only in D rows that are never stored.
//  - B out-of-bounds taps are zeroed by a per-tap float mask (tap-invariant),
//    loads come from clamped (always-valid) addresses.
// ---------------------------------------------------------------------------
__global__ __launch_bounds__(32) void conv1_wmma_kernel(
    const float* __restrict__ ctrl,   // (8,64,32,32)
    const float* __restrict__ w1,     // (50,64,3,3)
    const float* __restrict__ b1,     // (50,)
    float* __restrict__ h_out)        // (8,50,32,32)
{
  const int bid   = blockIdx.x;         // B * 4 * 64 = 2048 blocks
  const int b     = bid >> 8;
  const int rem   = bid & 255;
  const int mTile = rem >> 6;           // 0..3  (co tile)
  const int nTile = rem & 63;           // 0..63 (position tile)

  const int lane = threadIdx.x;
  const int half = lane >> 4;           // K-half selector
  const int lm   = lane & 15;
  const int kk   = half << 1;           // 0 or 2

  const int co  = mTile * 16 + lm;               // A-operand row
  const int coc = co < C1 ? co : 0;              // clamped address; garbage rows dropped

  const int pos = nTile * 16 + lm;               // B-operand column
  const int y   = pos >> 5;
  const int x   = pos & 31;

  v8f acc = {};

  for (int tap = 0; tap < 9; ++tap) {
    const int ky = tap / 3, kx = tap % 3;
    const int yy = y + ky - 1, xx = x + kx - 1;
    const bool inb = ((unsigned)yy < (unsigned)NHW) && ((unsigned)xx < (unsigned)NHW);
    const float bmask = inb ? 1.0f : 0.0f;
    const int yyc = inb ? yy : 0;
    const int xxc = inb ? xx : 0;

    const float* ap = w1 + coc * K1 + tap;                        // stride 9 in ci
    const float* bp = ctrl + ((b * CC) * NHW + yyc) * NHW + xxc;  // stride 1024 in ci

#pragma unroll
    for (int ci0 = 0; ci0 < CC; ci0 += 4) {
      const int ci = ci0 + kk;
      v2f A, Bv;
      A.x  = ap[ci * 9];
      A.y  = ap[(ci + 1) * 9];
      Bv.x = bp[ci * NPOS] * bmask;
      Bv.y = bp[(ci + 1) * NPOS] * bmask;
      acc = __builtin_amdgcn_wmma_f32_16x16x4_f32(
          false, A, false, Bv, (short)0, acc, false, false);
    }
  }

#pragma unroll
  for (int r = 0; r < 8; ++r) {
    const int coOut = mTile * 16 + half * 8 + r;   // D layout: M = r + 8*half
    if (coOut < C1) {
      float v = acc[r] + b1[coOut];
      h_out[(b * C1 + coOut) * NPOS + pos] = v > 0.0f ? v : 0.0f;
    }
  }
}

// ---------------------------------------------------------------------------
// Kernel 2: BatchNorm statistics per channel (population variance, ddof=0),
// folded into scale/shift:  xn*g + be == x*scale + shift.
// ---------------------------------------------------------------------------
__global__ __launch_bounds__(256) void bnstats_kernel(
    const float* __restrict__ h,      // (8,50,32,32)
    const float* __restrict__ gamma,
    const float* __restrict__ beta,
    float* __restrict__ scale,
    float* __restrict__ shift)
{
  __shared__ float ssum[256];
  __shared__ float ssq[256];
  const int ch  = blockIdx.x;          // 0..49
  const int tid = threadIdx.x;

  float s = 0.0f, q = 0.0f;
  for (int i = tid; i < BB * NPOS; i += 256) {
    const int b = i >> 10;
    const int p = i & (NPOS - 1);
    const float v = h[(b * C1 + ch) * NPOS + p];
    s += v;
    q += v * v;
  }
  ssum[tid] = s;
  ssq[tid]  = q;
  __syncthreads();
  for (int off = 128; off > 0; off >>= 1) {
    if (tid < off) {
      ssum[tid] += ssum[tid + off];
      ssq[tid]  += ssq[tid + off];
    }
    __syncthreads();
  }
  if (tid == 0) {
    const float inv_n = 1.0f / (float)(BB * NPOS);
    const float mean  = ssum[0] * inv_n;
    const float var   = ssq[0] * inv_n - mean * mean;
    const float rstd  = rsqrtf(var + 1e-5f);
    const float sc    = gamma[ch] * rstd;
    scale[ch] = sc;
    shift[ch] = beta[ch] - mean * sc;
  }
}

// ---------------------------------------------------------------------------
// Kernel 3: conv3x3 (50 -> 9) on the normalized tensor, WMMA implicit GEMM.
// BN folded on the fly into the B operand:  hn = h*scale + shift, with the
// scale/shift table zero-extended to 52 entries in LDS so the K padding
// (ci = 50,51) contributes exact zeros with no compares in the hot loop.
// SAME padding of the normalized input handled by the per-tap float mask.
// A rows m >= 9 are garbage but never stored.
// ---------------------------------------------------------------------------
__global__ __launch_bounds__(32) void conv2_wmma_kernel(
    const float* __restrict__ h,      // (8,50,32,32)
    const float* __restrict__ scale,  // (50,)
    const float* __restrict__ shift,  // (50,)
    const float* __restrict__ w2,     // (9,50,3,3)
    const float* __restrict__ b2,     // (9,)
    float* __restrict__ wts)          // (8,9,32,32)
{
  __shared__ float s_scale[52];
  __shared__ float s_shift[52];

  const int bid   = blockIdx.x;        // B * 64 = 512 blocks
  const int b     = bid >> 6;
  const int nTile = bid & 63;

  const int lane = threadIdx.x;
  const int half = lane >> 4;
  const int lm   = lane & 15;
  const int kk   = half << 1;

  for (int i = lane; i < 52; i += 32) {
    s_scale[i] = (i < C1) ? scale[i] : 0.0f;
    s_shift[i] = (i < C1) ? shift[i] : 0.0f;
  }
  __syncthreads();

  const int co  = lm;                  // only 0..8 valid
  const int coc = co < C2 ? co : 0;

  const int pos = nTile * 16 + lm;
  const int y   = pos >> 5;
  const int x   = pos & 31;

  v8f acc = {};

  for (int tap = 0; tap < 9; ++tap) {
    const int ky = tap / 3, kx = tap % 3;
    const int yy = y + ky - 1, xx = x + kx - 1;
    const bool inb = ((unsigned)yy < (unsigned)NHW) && ((unsigned)xx < (unsigned)NHW);
    const float bmask = inb ? 1.0f : 0.0f;
    const int yyc = inb ? yy : 0;
    const int xxc = inb ? xx : 0;

    const float* ap = w2 + coc * (C1 * 9) + tap;              // stride 9 in ci
    const float* bp = h + ((b * C1) * NHW + yyc) * NHW + xxc; // stride 1024 in ci

#pragma unroll
    for (int ci0 = 0; ci0 < 52; ci0 += 4) {                    // 50 padded to 52
      const int ci = ci0 + kk;
      const int c0 = ci < C1 ? ci : 0;          // clamped global addresses
      const int c1 = (ci + 1) < C1 ? (ci + 1) : 0;
      v2f A, Bv;
      A.x  = ap[c0 * 9];                        // garbage k>=50 zeroed via B
      A.y  = ap[c1 * 9];
      Bv.x = (bp[c0 * NPOS] * s_scale[ci]     + s_shift[ci])     * bmask;
      Bv.y = (bp[c1 * NPOS] * s_scale[ci + 1] + s_shift[ci + 1]) * bmask;
      acc = __builtin_amdgcn_wmma_f32_16x16x4_f32(
          false, A, false, Bv, (short)0, acc, false, false);
    }
  }

#pragma unroll
  for (int r = 0; r < 8; ++r) {
    const int coOut = half * 8 + r;
    if (coOut < C2) {
      wts[(b * C2 + coOut) * NPOS + pos] = acc[r] + b2[coOut];
    }
  }
}

// ---------------------------------------------------------------------------
// Kernel 4 (bandwidth-dominant): per-cell 9-way blend of shifted image views.
// One 256-thread block per (b, cell); 48x48 halo tile staged in LDS per
// channel (row stride 48 words over 64 banks -> conflict-free reads).
// out[b,c,h,w] = sum_{i,j} wts[b, i*3+j, h/16, w/16] * imgZ[b,c, h+(j-1)*16, w+(i-1)*16]
// ---------------------------------------------------------------------------
__global__ __launch_bounds__(256) void blend_kernel(
    const float* __restrict__ img,    // (8,3,512,512)
    const float* __restrict__ wts,    // (8,9,32,32)
    float* __restrict__ out)          // (8,3,512,512)
{
  __shared__ float tile[48 * 48];
  __shared__ float wk[9];

  const int bid = blockIdx.x;          // b*1024 + cy*32 + cx
  const int b   = bid >> 10;
  const int cy  = (bid >> 5) & 31;
  const int cx  = bid & 31;
  const int tid = threadIdx.x;
  const int ty  = tid >> 4;
  const int tx  = tid & 15;

  if (tid < 9) wk[tid] = wts[(b * C2 + tid) * NPOS + cy * NHW + cx];

  const int oy  = cy * CELLN + ty;
  const int ox  = cx * CELLN + tx;
  const int gy0 = cy * CELLN - CELLN;
  const int gx0 = cx * CELLN - CELLN;

  for (int ch = 0; ch < CIMG; ++ch) {
    const float* ip = img + (size_t)(b * CIMG + ch) * (HH * WW);

    for (int i = tid; i < 48 * 48; i += 256) {
      const int r  = i / 48;
      const int q  = i - r * 48;
      const int gy = gy0 + r;
      const int gx = gx0 + q;
      float v = 0.0f;
      if ((unsigned)gy < (unsigned)HH && (unsigned)gx < (unsigned)WW)
        v = ip[gy * WW + gx];
      tile[i] = v;
    }
    __syncthreads();

    float a = 0.0f;
#pragma unroll
    for (int i3 = 0; i3 < 3; ++i3) {
#pragma unroll
      for (int j3 = 0; j3 < 3; ++j3) {
        a += wk[i3 * 3 + j3] * tile[(ty + j3 * CELLN) * 48 + (tx + i3 * CELLN)];
      }
    }
    out[((size_t)(b * CIMG + ch) * HH + oy) * WW + ox] = a;
    __syncthreads();
  }
}

// ---------------------------------------------------------------------------
// Launch
// ---------------------------------------------------------------------------
extern "C" void kernel_launch(void* const* d_in, const int* in_sizes, int n_in,
                              void* d_out, int out_size, void* d_ws, size_t ws_size,
                              hipStream_t stream) {
  const float* img   = (const float*)d_in[0];
  const float* ctrl  = (const float*)d_in[1];
  const float* w1    = (const float*)d_in[2];
  const float* b1    = (const float*)d_in[3];
  const float* gamma = (const float*)d_in[4];
  const float* beta  = (const float*)d_in[5];
  const float* w2    = (const float*)d_in[6];
  const float* b2    = (const float*)d_in[7];
  float* out = (float*)d_out;

  float* ws    = (float*)d_ws;
  float* h     = ws;                       // 8*50*1024   = 409600 floats
  float* scale = ws + 409600;              // 50 floats
  float* shift = ws + 409664;              // 50 floats (64-aligned slot)
  float* wts   = ws + 409728;              // 8*9*1024    = 73728 floats

  conv1_wmma_kernel<<<BB * 4 * 64, 32, 0, stream>>>(ctrl, w1, b1, h);
  bnstats_kernel<<<C1, 256, 0, stream>>>(h, gamma, beta, scale, shift);
  conv2_wmma_kernel<<<BB * 64, 32, 0, stream>>>(h, scale, shift, w2, b2, wts);
  blend_kernel<<<BB * NHW * NHW, 256, 0, stream>>>(img, wts, out);
}